// GINLayer_20942260536132
// MI455X (gfx1250) — compile-verified
//
#include <hip/hip_runtime.h>

typedef __attribute__((ext_vector_type(2))) float v2f;
typedef __attribute__((ext_vector_type(8))) float v8f;

#define N_NODES 100000
#define DIM     128
#define N_EDGES 640000
#define BN_EPS  1e-5f

// ---------------------------------------------------------------------------
// Workspace layout (floats):
//   [0,            12.8M)  agg   : (1+eps)*x + scatter-add           51.2 MB
//   [12.8M,        25.6M)  y1    : agg @ w1 + b1 (pre-BN)            51.2 MB
//   [25.6M, 25.6M+1024)    stats : sum1,sq1,scale1,shift1,sum2,sq2,scale2,shift2
// ---------------------------------------------------------------------------
#define AGG_OFF   0
#define Y1_OFF    (N_NODES * DIM)
#define STATS_OFF (2 * N_NODES * DIM)

__global__ void zero_stats_kernel(float* __restrict__ stats) {
    stats[threadIdx.x] = 0.0f;           // 1 block x 1024 threads
}

__global__ void init_agg_kernel(const float4* __restrict__ x4,
                                const float* __restrict__ eps,
                                float4* __restrict__ agg4, int n4) {
    int i = blockIdx.x * blockDim.x + threadIdx.x;
    if (i >= n4) return;
    float s = 1.0f + eps[0];
    float4 v = x4[i];
    v.x *= s; v.y *= s; v.z *= s; v.w *= s;
    agg4[i] = v;
}

// One group of 32 threads per edge; each lane moves 4 floats (float4 gather,
// 4x global_atomic_add_f32 scatter). x and agg both fit in the 192MB L2.
__global__ void scatter_kernel(const float* __restrict__ x,
                               const int* __restrict__ row,
                               const int* __restrict__ col,
                               float* __restrict__ agg) {
    int i = blockIdx.x * blockDim.x + threadIdx.x;
    if (i >= N_EDGES * 32) return;
    int e = i >> 5;
    int q = (i & 31) << 2;                       // column offset 0..124
    int src = col[e];
    int dst = row[e];
    const float4 v = *(const float4*)(x + (size_t)src * DIM + q);
    float* d = agg + (size_t)dst * DIM + q;
    atomicAdd(d + 0, v.x);
    atomicAdd(d + 1, v.y);
    atomicAdd(d + 2, v.z);
    atomicAdd(d + 3, v.w);
}

// ---------------------------------------------------------------------------
// FP32 WMMA GEMM: C[100000x128] = op(A)[100000x128] @ B[128x128] + bias
// One wave per 16x16 output tile, 8 waves/block cover all column tiles of a
// 16-row strip. K loop: 32 x V_WMMA_F32_16X16X4_F32.
// FUSE_BN_RELU: A element -> relu(a*scale[k] + shift[k]) on the fly.
// ---------------------------------------------------------------------------
template <bool FUSE_BN_RELU>
__global__ void __launch_bounds__(256)
gemm_wmma_kernel(const float* __restrict__ A, const float* __restrict__ B,
                 const float* __restrict__ bias,
                 const float* __restrict__ scale,
                 const float* __restrict__ shift,
                 float* __restrict__ C) {
    const int wave = threadIdx.x >> 5;
    const int lane = threadIdx.x & 31;
    const int row0 = blockIdx.x << 4;       // 16-row strip
    const int n0   = wave << 4;             // 16-col tile (8 tiles = 128 cols)
    const int m    = lane & 15;             // A row in tile / B-C col in tile
    const int hk   = (lane >> 4) << 1;      // K sub-offset: 0 (lanes 0-15) or 2

    v8f acc = {};
    const float* arow = A + (size_t)(row0 + m) * DIM;

    for (int k0 = 0; k0 < DIM; k0 += 4) {
        const int k = k0 + hk;
        // A fragment: lane holds A[row0+m][k], A[row0+m][k+1]  (8B NT load)
        v2f a = __builtin_nontemporal_load((const v2f*)(arow + k));
        if (FUSE_BN_RELU) {
            float s0 = scale[k],     t0 = shift[k];
            float s1 = scale[k + 1], t1 = shift[k + 1];
            a.x = fmaxf(fmaf(a.x, s0, t0), 0.0f);
            a.y = fmaxf(fmaf(a.y, s1, t1), 0.0f);
        }
        // B fragment: lane holds B[k][n0+m], B[k+1][n0+m]  (hot in L0/L2)
        v2f b;
        b.x = B[(size_t)k * DIM + n0 + m];
        b.y = B[(size_t)(k + 1) * DIM + n0 + m];
        acc = __builtin_amdgcn_wmma_f32_16x16x4_f32(
            /*neg_a=*/false, a, /*neg_b=*/false, b,
            /*c_mod=*/(short)0, acc, /*reuse_a=*/false, /*reuse_b=*/false);
    }

    const float bv = bias[n0 + m];
    const int mbase = row0 + ((lane >> 4) << 3);   // +8 rows for upper lanes
    float* crow = C + (size_t)mbase * DIM + n0 + m;
#pragma unroll
    for (int r = 0; r < 8; ++r)
        crow[(size_t)r * DIM] = acc[r] + bv;
}

// Column sums / sums-of-squares over the node dim (for batch stats).
__global__ void __launch_bounds__(256)
colstats_kernel(const float* __restrict__ y,
                float* __restrict__ sum, float* __restrict__ sq) {
    __shared__ float ls[256];
    __shared__ float lq[256];
    const int c    = threadIdx.x & 127;
    const int half = threadIdx.x >> 7;
    float s = 0.0f, q = 0.0f;
    for (int r = blockIdx.x * 2 + half; r < N_NODES; r += gridDim.x * 2) {
        float v = y[(size_t)r * DIM + c];
        s += v;
        q += v * v;
    }
    ls[threadIdx.x] = s;
    lq[threadIdx.x] = q;
    __syncthreads();
    if (half == 0) {
        atomicAdd(&sum[c], ls[threadIdx.x] + ls[threadIdx.x + 128]);
        atomicAdd(&sq[c],  lq[threadIdx.x] + lq[threadIdx.x + 128]);
    }
}

// scale/shift per column: out = in*scale + shift  ==  (in-mean)*rsqrt(var+eps)*g + b
__global__ void bnparams_kernel(const float* __restrict__ sum,
                                const float* __restrict__ sq,
                                const float* __restrict__ gamma,
                                const float* __restrict__ beta,
                                float* __restrict__ scale,
                                float* __restrict__ shift) {
    int c = threadIdx.x;
    if (c >= DIM) return;
    const float inv_n = 1.0f / (float)N_NODES;
    float mean = sum[c] * inv_n;
    float var  = sq[c] * inv_n - mean * mean;   // biased batch variance
    float sc   = gamma[c] * rsqrtf(var + BN_EPS);
    scale[c] = sc;
    shift[c] = beta[c] - mean * sc;
}

// Final in-place BN2 + ReLU on d_out.
__global__ void bnrelu_kernel(float4* __restrict__ y,
                              const float* __restrict__ scale,
                              const float* __restrict__ shift, int n4) {
    int i = blockIdx.x * blockDim.x + threadIdx.x;
    if (i >= n4) return;
    int c = (i << 2) & 127;
    float4 v = y[i];
    v.x = fmaxf(fmaf(v.x, scale[c + 0], shift[c + 0]), 0.0f);
    v.y = fmaxf(fmaf(v.y, scale[c + 1], shift[c + 1]), 0.0f);
    v.z = fmaxf(fmaf(v.z, scale[c + 2], shift[c + 2]), 0.0f);
    v.w = fmaxf(fmaf(v.w, scale[c + 3], shift[c + 3]), 0.0f);
    y[i] = v;
}

extern "C" void kernel_launch(void* const* d_in, const int* in_sizes, int n_in,
                              void* d_out, int out_size, void* d_ws, size_t ws_size,
                              hipStream_t stream) {
    // setup_inputs order: x, edge_index, eps, w1, b1, gamma1, beta1, w2, b2, gamma2, beta2
    const float* x          = (const float*)d_in[0];
    const int*   edge_index = (const int*)d_in[1];
    const float* eps        = (const float*)d_in[2];
    const float* w1         = (const float*)d_in[3];
    const float* b1         = (const float*)d_in[4];
    const float* gamma1     = (const float*)d_in[5];
    const float* beta1      = (const float*)d_in[6];
    const float* w2         = (const float*)d_in[7];
    const float* b2         = (const float*)d_in[8];
    const float* gamma2     = (const float*)d_in[9];
    const float* beta2      = (const float*)d_in[10];

    const int* e_row = edge_index;            // edge_index[0]
    const int* e_col = edge_index + N_EDGES;  // edge_index[1]

    float* ws    = (float*)d_ws;
    float* agg   = ws + AGG_OFF;
    float* y1    = ws + Y1_OFF;
    float* stats = ws + STATS_OFF;
    float* sum1 = stats +   0; float* sq1    = stats + 128;
    float* sc1  = stats + 256; float* sh1    = stats + 384;
    float* sum2 = stats + 512; float* sq2    = stats + 640;
    float* sc2  = stats + 768; float* sh2    = stats + 896;

    float* out = (float*)d_out;

    const int n_elems = N_NODES * DIM;        // 12,800,000
    const int n4      = n_elems / 4;          //  3,200,000

    // 1) zero BN accumulators (graph-replay safe: re-zeroed every call)
    zero_stats_kernel<<<1, 1024, 0, stream>>>(stats);

    // 2) agg = (1+eps) * x
    init_agg_kernel<<<(n4 + 255) / 256, 256, 0, stream>>>(
        (const float4*)x, eps, (float4*)agg, n4);

    // 3) scatter-add neighbor features
    scatter_kernel<<<(N_EDGES * 32 + 255) / 256, 256, 0, stream>>>(
        x, e_row, e_col, agg);

    // 4) y1 = agg @ w1 + b1   (fp32 WMMA)
    gemm_wmma_kernel<false><<<N_NODES / 16, 256, 0, stream>>>(
        agg, w1, b1, nullptr, nullptr, y1);

    // 5-6) batch stats + affine fold for BN1
    colstats_kernel<<<1024, 256, 0, stream>>>(y1, sum1, sq1);
    bnparams_kernel<<<1, 128, 0, stream>>>(sum1, sq1, gamma1, beta1, sc1, sh1);

    // 7) out_pre = relu(bn1(y1)) @ w2 + b2   (BN1+ReLU fused into A operand)
    gemm_wmma_kernel<true><<<N_NODES / 16, 256, 0, stream>>>(
        y1, w2, b2, sc1, sh1, out);

    // 8-9) batch stats + affine fold for BN2
    colstats_kernel<<<1024, 256, 0, stream>>>(out, sum2, sq2);
    bnparams_kernel<<<1, 128, 0, stream>>>(sum2, sq2, gamma2, beta2, sc2, sh2);

    // 10) out = relu(bn2(out)) in place
    bnrelu_kernel<<<(n4 + 255) / 256, 256, 0, stream>>>(
        (float4*)out, sc2, sh2, n4);
}